// CategoricalCrossEntropy_80358838108292
// MI455X (gfx1250) — compile-verified
//
#include <hip/hip_runtime.h>

typedef __attribute__((ext_vector_type(2))) float v2f;
typedef __attribute__((ext_vector_type(4))) float v4f;
typedef __attribute__((ext_vector_type(8))) float v8f;

#define CE_EPS 1e-8f

// ---------------------------------------------------------------------------
// Wave32 full reduction of one float per lane -> total broadcast to all lanes.
// Uses V_WMMA_F32_16X16X4_F32 with B = ones(4x16):
//   A layout (16x4 f32): lanes 0-15 -> A[m,0]=a.x, A[m,1]=a.y
//                        lanes16-31 -> A[m,2]=a.x, A[m,3]=a.y
//   With a = {p_lane, 0}:  D[m,n] = p_m + p_{m+16}  (independent of n)
//   D layout: VGPR r holds M=r (lanes 0-15) / M=r+8 (lanes 16-31)
//   => sum(d[0..7]) gives half-total per 16-lane group; SWAPX16 swizzle adds
//      the two halves. EXEC must be all-1s at the call site.
// ---------------------------------------------------------------------------
__device__ __forceinline__ float wave_reduce_full(float p) {
#if __has_builtin(__builtin_amdgcn_wmma_f32_16x16x4_f32)
  v2f a; a.x = p; a.y = 0.0f;
  v2f b; b.x = 1.0f; b.y = 1.0f;
  v8f c = {};
  v8f d = __builtin_amdgcn_wmma_f32_16x16x4_f32(
      /*neg_a=*/false, a, /*neg_b=*/false, b,
      /*c_mod=*/(short)0, c, /*reuse_a=*/false, /*reuse_b=*/false);
  float s = ((d[0] + d[1]) + (d[2] + d[3])) + ((d[4] + d[5]) + (d[6] + d[7]));
  // cross-half add: SWAPX16 (group-of-32 swizzle: xor=0x10, and=0x1f)
  int sw = __builtin_amdgcn_ds_swizzle(__float_as_int(s), 0x401F);
  return s + __int_as_float(sw);
#else
  // Fallback: xor-swizzle tree (still wave32-correct)
  float s = p;
  s += __int_as_float(__builtin_amdgcn_ds_swizzle(__float_as_int(s), 0x041F)); // xor 1
  s += __int_as_float(__builtin_amdgcn_ds_swizzle(__float_as_int(s), 0x081F)); // xor 2
  s += __int_as_float(__builtin_amdgcn_ds_swizzle(__float_as_int(s), 0x101F)); // xor 4
  s += __int_as_float(__builtin_amdgcn_ds_swizzle(__float_as_int(s), 0x201F)); // xor 8
  s += __int_as_float(__builtin_amdgcn_ds_swizzle(__float_as_int(s), 0x401F)); // xor 16
  return s;
#endif
}

// ---------------------------------------------------------------------------
// Pass 1: grid-stride over float4 pairs; per-thread fp32 accumulate of
// input * log(target + eps); WMMA wave reduce; LDS cross-wave; one partial
// per block. Non-temporal B128 loads (268 MB stream, no reuse).
// ---------------------------------------------------------------------------
__global__ void __launch_bounds__(256)
ce_partial_kernel(const float* __restrict__ inp, const float* __restrict__ tgt,
                  float* __restrict__ partials, long long n4) {
  long long tid    = (long long)blockIdx.x * blockDim.x + threadIdx.x;
  long long stride = (long long)gridDim.x * blockDim.x;
  const v4f* __restrict__ ip = (const v4f*)inp;
  const v4f* __restrict__ tp = (const v4f*)tgt;

  float acc = 0.0f;
  for (long long i = tid; i < n4; i += stride) {
    v4f x = __builtin_nontemporal_load(ip + i);
    v4f t = __builtin_nontemporal_load(tp + i);
    acc = fmaf(x.x, __logf(t.x + CE_EPS), acc);
    acc = fmaf(x.y, __logf(t.y + CE_EPS), acc);
    acc = fmaf(x.z, __logf(t.z + CE_EPS), acc);
    acc = fmaf(x.w, __logf(t.w + CE_EPS), acc);
  }

  // EXEC is all-1s here (trip count uniform across the wave when n4 is a
  // multiple of total threads; loop-exit is per-lane-identical regardless
  // since consecutive lanes have consecutive i).
  float wsum = wave_reduce_full(acc);

  __shared__ float lds[8];
  int lane = threadIdx.x & 31;
  int wave = threadIdx.x >> 5;
  if (lane == 0) lds[wave] = wsum;
  __syncthreads();
  if (threadIdx.x == 0) {
    int nw = blockDim.x >> 5;
    float b = 0.0f;
    for (int w = 0; w < nw; ++w) b += lds[w];
    partials[blockIdx.x] = b;
  }
}

// ---------------------------------------------------------------------------
// Pass 2: single block reduces the block partials deterministically and
// writes out[0] = -sum / B.  nparts is a multiple of blockDim.x so the loop
// trip count is uniform -> EXEC stays all-1s at the WMMA.
// ---------------------------------------------------------------------------
__global__ void __launch_bounds__(256)
ce_final_kernel(const float* __restrict__ partials, int nparts,
                float* __restrict__ out, float invB) {
  float acc = 0.0f;
  for (int i = threadIdx.x; i < nparts; i += blockDim.x)
    acc += partials[i];

  float wsum = wave_reduce_full(acc);

  __shared__ float lds[8];
  int lane = threadIdx.x & 31;
  int wave = threadIdx.x >> 5;
  if (lane == 0) lds[wave] = wsum;
  __syncthreads();
  if (threadIdx.x == 0) {
    int nw = blockDim.x >> 5;
    float b = 0.0f;
    for (int w = 0; w < nw; ++w) b += lds[w];
    out[0] = -b * invB;
  }
}

extern "C" void kernel_launch(void* const* d_in, const int* in_sizes, int n_in,
                              void* d_out, int out_size, void* d_ws, size_t ws_size,
                              hipStream_t stream) {
  const float* inp = (const float*)d_in[0];
  const float* tgt = (const float*)d_in[1];
  float* out      = (float*)d_out;
  float* partials = (float*)d_ws;

  long long n  = (long long)in_sizes[0];   // B*C elements (33,554,432)
  long long n4 = n >> 2;                   // float4 count (C=128 => divisible)

  const int NT = 256;
  int NB = 2048;                           // 16 float4 iters/thread/array
  // keep NB a multiple of NT so pass-2's loop trip count is wave-uniform,
  // and never exceed workspace capacity
  size_t maxParts = ws_size / sizeof(float);
  while ((size_t)NB > maxParts && NB > NT) NB >>= 1;

  float invB = 128.0f / (float)n;          // 1/B with C == 128

  ce_partial_kernel<<<NB, NT, 0, stream>>>(inp, tgt, partials, n4);
  ce_final_kernel<<<1, NT, 0, stream>>>(partials, NB, out, invB);
}